// InvConditionalUnitriLinear_17961553232466
// MI455X (gfx1250) — compile-verified
//
#include <hip/hip_runtime.h>
#include <hip/hip_bf16.h>

// Problem constants (from reference): x[32,8,128,128], w[32,28,128,128] -> z[32,8,128,128]
#define Dch   8
#define C28   28
#define HWc   16384          // 128*128
#define NIMG  32
#define TP    128            // pixels per tile (= threads per block)
#define TILES_PER_IMG (HWc / TP)          // 128
#define NTILES        (NIMG * TILES_PER_IMG)   // 4096
#define NBLOCKS 1024         // NTILES % NBLOCKS == 0 -> 4 tiles per block, pipelined

typedef unsigned int v4u __attribute__((ext_vector_type(4)));
typedef int          v4i __attribute__((ext_vector_type(4)));
typedef int          v8i __attribute__((ext_vector_type(8)));
typedef float        vf4 __attribute__((ext_vector_type(4)));

#if __has_builtin(__builtin_amdgcn_tensor_load_to_lds) && __has_builtin(__builtin_amdgcn_s_wait_tensorcnt)
#define USE_TDM 1
#endif

#ifdef USE_TDM

// Issue one 2-D TDM descriptor: gather `rows` channel planes (row stride = row_stride
// elements) of `tp` contiguous pixels each into LDS at byte offset lds_off.
// D# layout per CDNA5 ISA ch.8 (group0 128b, group1 256b; groups 2/3 zero => 2-D).
__device__ __forceinline__ void tdm_load_2d(unsigned lds_off, const float* gptr,
                                            unsigned rows, unsigned row_stride,
                                            unsigned tp) {
  unsigned long long ga = (unsigned long long)(size_t)gptr;
  v4u g0;
  g0[0] = 1u;                                        // count=1 (valid), user mode
  g0[1] = lds_off;                                   // lds_addr (bytes)
  g0[2] = (unsigned)(ga & 0xFFFFFFFFu);              // global_addr[31:0]
  g0[3] = (unsigned)((ga >> 32) & 0x1FFFFFFu)        // global_addr[56:32]
        | (2u << 30);                                // type = 2 ("image")
  v8i g1;
  g1[0] = (int)(2u << 16);                           // data_size=2 (4 bytes); mask/flags=0
  g1[1] = (int)(tp << 16);                           // tensor_dim0[15:0] << 16 (barrier addr 0)
  g1[2] = (int)((tp >> 16) | (rows << 16));          // tensor_dim0[31:16] | tensor_dim1[15:0]
  g1[3] = (int)((rows >> 16) | (tp << 16));          // tensor_dim1[31:16] | tile_dim0
  g1[4] = (int)rows;                                 // tile_dim1 (tile_dim2 = 0)
  g1[5] = (int)row_stride;                           // tensor_dim0_stride[31:0]
  g1[6] = 0;                                         // stride hi | dim1_stride lo (unused 2-D)
  g1[7] = 0;
  v4i z = {};
#if defined(__clang_major__) && (__clang_major__ >= 23)
  v8i z8 = {};
  __builtin_amdgcn_tensor_load_to_lds(g0, g1, z, z, z8, 0);
#else
  __builtin_amdgcn_tensor_load_to_lds(g0, g1, z, z, 0);
#endif
}

__global__ void __launch_bounds__(TP, 1)
unitri_solve_tdm(const float* __restrict__ x, const float* __restrict__ w,
                 float* __restrict__ out) {
  __shared__ float sX[2][Dch * TP];   //  2 *  8 * 128 * 4 =  8 KB
  __shared__ float sW[2][C28 * TP];   //  2 * 28 * 128 * 4 = 28 KB  (36 KB total)

  const int tid   = threadIdx.x;
  const bool wave0 = (tid >> 5) == 0;

  auto issue = [&](int t, int b) {
    int n  = t / TILES_PER_IMG;
    int ts = (t % TILES_PER_IMG) * TP;
    const float* xg = x + (size_t)n * Dch * HWc + ts;
    const float* wg = w + (size_t)n * C28 * HWc + ts;
    tdm_load_2d((unsigned)(size_t)&sX[b][0], xg, Dch, HWc, TP);
    tdm_load_2d((unsigned)(size_t)&sW[b][0], wg, C28, HWc, TP);
  };

  const int t0 = blockIdx.x;           // grid-stride over tiles; gridDim.x <= NTILES
  if (wave0) issue(t0, 0);             // prime the pipeline (buffer 0)

  int it = 0;
  for (int t = t0; t < NTILES; t += NBLOCKS, ++it) {
    const int nt = t + NBLOCKS;
    if (wave0) {
      if (nt < NTILES) {
        issue(nt, (it + 1) & 1);                    // prefetch next tile -> other buffer
        __builtin_amdgcn_s_wait_tensorcnt(2);       // in-order: current tile's 2 ops done
      } else {
        __builtin_amdgcn_s_wait_tensorcnt(0);       // drain
      }
    }
    __syncthreads();                                 // LDS tile visible to all waves

    const int b = it & 1;
    float z[Dch];
#pragma unroll
    for (int c = 0; c < Dch; ++c) z[c] = sX[b][c * TP + tid];
    // unit upper-triangular solve, column sweep: z[i] -= W[i,j] * z[j], j = 7..1
#pragma unroll
    for (int j = Dch - 1; j >= 1; --j) {
      const int base = j * (j - 1) / 2;
#pragma unroll
      for (int i = 0; i < j; ++i)
        z[i] = fmaf(-sW[b][(base + i) * TP + tid], z[j], z[i]);
    }

    const int n  = t / TILES_PER_IMG;
    const int ts = (t % TILES_PER_IMG) * TP;
    float* og = out + (size_t)n * Dch * HWc + ts + tid;
#pragma unroll
    for (int c = 0; c < Dch; ++c)
      __builtin_nontemporal_store(z[c], og + c * HWc);   // streaming: bypass-friendly TH

    __syncthreads();                                 // buffer b safe to overwrite next iter
  }
}

#endif // USE_TDM

// Fallback / reference path: direct streaming kernel, float4 (b128) per thread,
// non-temporal loads+stores, grid-stride. Bandwidth-optimal without TDM.
__global__ void __launch_bounds__(256)
unitri_solve_direct(const float* __restrict__ x, const float* __restrict__ w,
                    float* __restrict__ out) {
  const int QPI   = HWc / 4;              // float4 quads per image plane
  const int total = NIMG * QPI;
  for (int q = blockIdx.x * blockDim.x + threadIdx.x; q < total;
       q += gridDim.x * blockDim.x) {
    const int n  = q / QPI;
    const int p4 = (q - n * QPI) * 4;
    const vf4* xb = (const vf4*)(x + (size_t)n * Dch * HWc + p4);
    const vf4* wb = (const vf4*)(w + (size_t)n * C28 * HWc + p4);
    vf4 z[Dch];
#pragma unroll
    for (int c = 0; c < Dch; ++c)
      z[c] = __builtin_nontemporal_load(&xb[(size_t)c * (HWc / 4)]);
#pragma unroll
    for (int j = Dch - 1; j >= 1; --j) {
      const int base = j * (j - 1) / 2;
#pragma unroll
      for (int i = 0; i < j; ++i) {
        vf4 wc = __builtin_nontemporal_load(&wb[(size_t)(base + i) * (HWc / 4)]);
#pragma unroll
        for (int k = 0; k < 4; ++k) z[i][k] = fmaf(-wc[k], z[j][k], z[i][k]);
      }
    }
    vf4* ob = (vf4*)(out + (size_t)n * Dch * HWc + p4);
#pragma unroll
    for (int c = 0; c < Dch; ++c)
      __builtin_nontemporal_store(z[c], &ob[(size_t)c * (HWc / 4)]);
  }
}

extern "C" void kernel_launch(void* const* d_in, const int* in_sizes, int n_in,
                              void* d_out, int out_size, void* d_ws, size_t ws_size,
                              hipStream_t stream) {
  const float* x = (const float*)d_in[0];   // [32, 8, 128, 128]
  const float* w = (const float*)d_in[1];   // [32, 28, 128, 128]
  float* out = (float*)d_out;               // [32, 8, 128, 128]
  (void)in_sizes; (void)n_in; (void)out_size; (void)d_ws; (void)ws_size;
#ifdef USE_TDM
  unitri_solve_tdm<<<NBLOCKS, TP, 0, stream>>>(x, w, out);
#else
  unitri_solve_direct<<<2048, 256, 0, stream>>>(x, w, out);
#endif
}